// Saturate_32040456028507
// MI455X (gfx1250) — compile-verified
//
#include <hip/hip_runtime.h>
#include <hip/hip_bf16.h>

typedef __attribute__((ext_vector_type(4))) float v4f;

#define HWV    65536u          // float4s per 512x512 plane
#define PLANE3 (3u * HWV)      // float4s per image (3 planes)
#define TPB    256u

__device__ __forceinline__ void adjust_pixel(float r, float g, float b,
                                             float& ro, float& go, float& bo) {
    float maxc  = fmaxf(fmaxf(r, g), b);
    float minc  = fminf(fminf(r, g), b);
    float delta = maxc - minc;
    bool  has_d = delta > 0.0f;
    bool  has_m = maxc  > 0.0f;

    float inv_d = __builtin_amdgcn_rcpf(has_d ? delta : 1.0f);
    float inv_m = __builtin_amdgcn_rcpf(has_m ? maxc  : 1.0f);

    // hue selection chain: priority maxc==r, then maxc==g, else b
    float h = (maxc == r) ? (g - b) * inv_d
            : (maxc == g) ? 2.0f + (b - r) * inv_d
                          : 4.0f + (r - g) * inv_d;
    h = h * (1.0f / 6.0f);
    h = has_d ? (h - floorf(h)) : 0.0f;          // (h/6) mod 1.0, sign of divisor

    float s = has_m ? delta * inv_m : 0.0f;
    s = fminf(fmaxf(s * 2.0f, 0.0f), 1.0f);      // clip(s*SAT_MUL + SAT_ADD, 0, 1)

    float v  = maxc;
    float h6 = h * 6.0f;                         // in [0, 6)
    float vs = v * s;

    // f(n): k = (n + h6) mod 6 ; out = v - v*s*clip(min(k, 4-k), 0, 1)
    auto f = [&](float n) {
        float x = n + h6;                        // in [1, 12): positive
        float k = x - 6.0f * floorf(x * (1.0f / 6.0f));
        float w = fminf(k, 4.0f - k);
        w = fminf(fmaxf(w, 0.0f), 1.0f);
        return v - vs * w;
    };
    ro = f(5.0f);
    go = f(3.0f);
    bo = f(1.0f);
}

__global__ void __launch_bounds__(TPB)
saturate_kernel(const float* __restrict__ in_base, v4f* __restrict__ out4) {
    __shared__ v4f tile[3 * TPB];                // 12 KB: R | G | B slots

    const unsigned t    = threadIdx.x;
    const unsigned v    = blockIdx.x * TPB + t;  // pixel-vec index
    const unsigned img  = v >> 16;               // v / HWV
    const unsigned offv = v & 0xFFFFu;           // v % HWV

    const unsigned rIdx = img * PLANE3 + offv;   // float4 indices
    const unsigned gIdx = rIdx + HWV;
    const unsigned bIdx = gIdx + HWV;

    const unsigned ldsR = (unsigned)(size_t)(void*)&tile[t];
    const unsigned ldsG = (unsigned)(size_t)(void*)&tile[TPB + t];
    const unsigned ldsB = (unsigned)(size_t)(void*)&tile[2 * TPB + t];

    // Async global -> LDS, 128b per lane, GVS addressing (saddr64 + vaddr32).
    asm volatile("global_load_async_to_lds_b128 %0, %1, %2"
                 :: "v"(ldsR), "v"(rIdx * 16u), "s"(in_base) : "memory");
    asm volatile("global_load_async_to_lds_b128 %0, %1, %2"
                 :: "v"(ldsG), "v"(gIdx * 16u), "s"(in_base) : "memory");
    asm volatile("global_load_async_to_lds_b128 %0, %1, %2"
                 :: "v"(ldsB), "v"(bIdx * 16u), "s"(in_base) : "memory");
    asm volatile("s_wait_asynccnt 0" ::: "memory");

    v4f R = tile[t];
    v4f G = tile[TPB + t];
    v4f B = tile[2 * TPB + t];

    v4f Ro, Go, Bo;
#pragma unroll
    for (int i = 0; i < 4; ++i) {
        float ro, go, bo;
        adjust_pixel(R[i], G[i], B[i], ro, go, bo);
        Ro[i] = ro;
        Go[i] = go;
        Bo[i] = bo;
    }

    // Write-once output: non-temporal 128b stores.
    __builtin_nontemporal_store(Ro, &out4[rIdx]);
    __builtin_nontemporal_store(Go, &out4[gIdx]);
    __builtin_nontemporal_store(Bo, &out4[bIdx]);
}

extern "C" void kernel_launch(void* const* d_in, const int* in_sizes, int n_in,
                              void* d_out, int out_size, void* d_ws, size_t ws_size,
                              hipStream_t stream) {
    const float* in  = (const float*)d_in[0];
    float*       out = (float*)d_out;

    // n = 64*3*512*512 floats; pixel-vecs = n / 3 / 4
    const unsigned n        = (unsigned)in_sizes[0];
    const unsigned npix_vec = n / 12u;           // 4,194,304
    const unsigned blocks   = npix_vec / TPB;    // 16,384

    saturate_kernel<<<dim3(blocks), dim3(TPB), 0, stream>>>(in, (v4f*)out);
}